// NITSPrimitive_47330539602027
// MI455X (gfx1250) — compile-verified
//
#include <hip/hip_runtime.h>
#include <hip/hip_bf16.h>

// CDNA5 / gfx1250, wave32. One wave per sample; 16x16 matvec via
// V_WMMA_F32_16X16X4_F32 accumulated over 4 K-chunks (exact f32 math).

typedef __attribute__((ext_vector_type(2))) float v2f;
typedef __attribute__((ext_vector_type(8))) float v8f;

__global__ __launch_bounds__(256) void nits_wmma_kernel(
    const float* __restrict__ x,       // [N]
    const float* __restrict__ params,  // [N, 320]
    float* __restrict__ out,           // [N]
    int N)
{
    const int lane = threadIdx.x & 31;
    const int wid  = threadIdx.x >> 5;           // 8 waves per block
    const int s    = blockIdx.x * 8 + wid;       // sample for this wave
    if (s >= N) return;                          // wave-uniform branch

    const float* __restrict__ P = params + (size_t)s * 320;
    const float xv = x[s];

    // ---------------- Layer 0 (16x1 matvec + bias + sigmoid) ----------------
    // One coalesced 128B load: P[0..15]=W0 rows, P[16..31]=b0.
    const float v0 = P[lane];
    const float a0 = __shfl(v0, lane & 15);
    const float b0 = __shfl(v0, (lane & 15) + 16);
    const float pre0 = __expf(fminf(a0, 7.0f)) * xv + b0;
    // lane j (and j+16) holds h0[j & 15]
    const float h = 1.0f / (1.0f + __expf(-pre0));

    // ---------------- Layer 1 (16x16 matvec) via WMMA f32 16x16x4 -----------
    // A layout (32-bit 16x4): lanes 0-15 -> M=lane, VGPR0/1 = K=0,1
    //                         lanes 16-31 -> M=lane-16, VGPR0/1 = K=2,3
    // B layout (32-bit 4x16): VGPR0 = rows K=0 (lanes 0-15) / K=2 (16-31),
    //                         VGPR1 = rows K=1 / K=3. Columns broadcast.
    const int m     = lane & 15;
    const int khalf = (lane >> 4) << 1;          // 0 or 2
    v8f acc = {};
    #pragma unroll
    for (int c = 0; c < 4; ++c) {
        // Two consecutive K entries of exp(W1[m][.]) for this lane.
        const float2 w = *(const float2*)(P + 32 + m * 16 + 4 * c + khalf);
        v2f a;
        a.x = __expf(fminf(w.x, 7.0f));
        a.y = __expf(fminf(w.y, 7.0f));
        const int src = 4 * c + khalf;
        v2f b;
        b.x = __shfl(h, src);
        b.y = __shfl(h, src + 1);
        acc = __builtin_amdgcn_wmma_f32_16x16x4_f32(
            /*neg_a=*/false, a, /*neg_b=*/false, b,
            /*c_mod=*/(short)0, acc, /*reuse_a=*/false, /*reuse_b=*/false);
    }
    // D: VGPR r = t[r] on lanes 0-15, t[r+8] on lanes 16-31 (replicated over N).
    // Gather t[lane&15] into each lane.
    float tj = 0.0f;
    const int srcl = (lane & 8) ? 16 : 0;
    #pragma unroll
    for (int r = 0; r < 8; ++r) {
        const float vv = __shfl(acc[r], srcl);
        tj = ((lane & 7) == r) ? vv : tj;
    }

    // ---------------- Layer 1 bias + sigmoid --------------------------------
    const float b1 = P[288 + m];
    const float h1 = 1.0f / (1.0f + __expf(-(tj + b1)));

    // ---------------- Layer 2: softmax(W2) . h1 -----------------------------
    const float r2 = P[304 + m];
    float mx = r2;
    #pragma unroll
    for (int o = 8; o >= 1; o >>= 1) mx = fmaxf(mx, __shfl_xor(mx, o));
    const float e = __expf(r2 - mx);
    float num = e * h1;
    float den = e;
    #pragma unroll
    for (int o = 8; o >= 1; o >>= 1) {
        num += __shfl_xor(num, o);
        den += __shfl_xor(den, o);
    }

    if (lane == 0) out[s] = num / den + 0.001f * xv;
}

extern "C" void kernel_launch(void* const* d_in, const int* in_sizes, int n_in,
                              void* d_out, int out_size, void* d_ws, size_t ws_size,
                              hipStream_t stream) {
    const float* x      = (const float*)d_in[0];  // [N] (N x 1)
    const float* params = (const float*)d_in[1];  // [N, 320]
    float* out          = (float*)d_out;          // [N]
    const int N = in_sizes[0];

    const int waves_per_block = 8;                // 256 threads
    const int blocks = (N + waves_per_block - 1) / waves_per_block;
    nits_wmma_kernel<<<blocks, 256, 0, stream>>>(x, params, out, N);
}